// PFSTLossV2_41472204210800
// MI455X (gfx1250) — compile-verified
//
#include <hip/hip_runtime.h>
#include <hip/hip_bf16.h>
#include <cstdint>

typedef __attribute__((ext_vector_type(2))) float v2f;
typedef __attribute__((ext_vector_type(8))) float v8f;
typedef __attribute__((ext_vector_type(4))) unsigned int v4u;
typedef __attribute__((ext_vector_type(4))) int v4i;
typedef __attribute__((ext_vector_type(8))) int v8i;

#define HB 2       // batch
#define NC 19      // classes
#define HT 256     // logits H == W
#define HF 128     // feature H == W
#define CF 256     // feature channels
#define INV_SIG2 (1.0f / 900.0f)

// TDM tile: 128 px (x) * 3 rows (y) * 16 ch (z), f32, LDS pad 8 DW every 128 DW
#define TDM_CH 16                    // channels per chunk
#define ROWSTR 136                   // LDS words per px-row (128 + 8 pad)
#define CHUNK_WORDS (TDM_CH * 3 * ROWSTR)   // 6528 words = 26112 B per buffer

// ---------------- prep: softmax prob + argmax pred ----------------
__global__ __launch_bounds__(256) void prep_kernel(const float* __restrict__ logits,
                                                   float* __restrict__ prob,
                                                   int* __restrict__ pred) {
  int idx = blockIdx.x * 256 + threadIdx.x;       // exactly HB*HT*HT threads
  int b   = idx >> 16;
  int pix = idx & 65535;
  const float* lg = logits + ((size_t)b * NC) * (HT * HT) + pix;
  float mx = -3.4e38f;
  int am = 0;
#pragma unroll
  for (int c = 0; c < NC; ++c) {
    float v = lg[(size_t)c * (HT * HT)];
    if (v > mx) { mx = v; am = c; }                // strict '>' keeps first max (jnp.argmax)
  }
  float e[NC];
  float s = 0.0f;
#pragma unroll
  for (int c = 0; c < NC; ++c) {
    float v = __expf(lg[(size_t)c * (HT * HT)] - mx);
    e[c] = v;
    s += v;
  }
  float inv = 1.0f / s;
  float* pr = prob + ((size_t)b * NC) * (HT * HT) + pix;
#pragma unroll
  for (int c = 0; c < NC; ++c) pr[(size_t)c * (HT * HT)] = e[c] * inv;
  pred[idx] = am;
}

// ---------------- feature L2 norms at 128x128 ----------------
__global__ __launch_bounds__(128) void norm_kernel(const float* __restrict__ x_ema,
                                                   const float* __restrict__ x_src,
                                                   float* __restrict__ norms) {
  int x = threadIdx.x;
  int y = blockIdx.x;
  int b = blockIdx.y;
  int inp = blockIdx.z;
  const float* src = (inp == 0 ? x_ema : x_src) + (size_t)b * CF * (HF * HF);
  float s = 0.0f;
  for (int c = 0; c < CF; ++c) {
    float v = src[(size_t)c * (HF * HF) + y * HF + x];
    s += v * v;
  }
  norms[((size_t)(inp * HB + b)) * (HF * HF) + y * HF + x] = s;
}

// Issue one TDM load: tile = 128 px x 3 rows x TDM_CH channels starting at gptr.
// Tensor dims == tile dims (tile is fully in-bounds after row clamping), so the
// TDM never reads OOB. pad_enable inserts 8 DWORDs after every 128 -> row
// stride ROWSTR, giving conflict-free ds_load_b32 WMMA fragment reads.
__device__ __forceinline__ void tdm_load_chunk(const float* gptr, unsigned lds_byte_off) {
  unsigned long long ga = (unsigned long long)(uintptr_t)gptr;
  v4u g0;
  g0.x = 1u;                                          // count=1, user desc
  g0.y = lds_byte_off;                                // lds_addr (bytes)
  g0.z = (unsigned)(ga & 0xffffffffu);                // global_addr[31:0]
  g0.w = (unsigned)((ga >> 32) & 0x01ffffffu) | (2u << 30);  // addr[56:32] | type=2
  v8i g1;
  g1[0] = (int)((2u << 16) |                          // data_size = 4 bytes
                (1u << 20) |                          // pad_enable
                (6u << 22) |                          // pad_interval: every 128 DW
                (7u << 25));                          // pad_amount: 8 DW
  g1[1] = (int)(128u << 16);                          // tensor_dim0 = 128 (lo16)
  g1[2] = (int)(3u << 16);                            // dim0 hi=0 | tensor_dim1 = 3
  g1[3] = (int)(128u << 16);                          // dim1 hi=0 | tile_dim0 = 128
  g1[4] = (int)(3u | ((unsigned)TDM_CH << 16));       // tile_dim1 = 3 | tile_dim2
  g1[5] = (int)128u;                                  // tensor_dim0_stride = 128 (row)
  g1[6] = (int)((unsigned)(HF * HF) << 16);           // stride0 hi=0 | dim1_stride lo16 = 16384
  g1[7] = 0;                                          // dim1_stride hi = 0
  v4i g2 = {TDM_CH, 0, 0, 0};                         // tensor_dim2 = TDM_CH
  v4i g3 = {0, 0, 0, 0};
#if __clang_major__ >= 23
  v8i gz = {0, 0, 0, 0, 0, 0, 0, 0};
  __builtin_amdgcn_tensor_load_to_lds(g0, g1, g2, g3, gz, 0);
#else
  __builtin_amdgcn_tensor_load_to_lds(g0, g1, g2, g3, 0);
#endif
}

// ---------------- 9-direction gaussian similarity via f32 WMMA + TDM ----------------
// Grid: (HF rows, HB batches, 2 inputs), block = 256 threads = 8 waves.
// Wave 0 double-buffers 16-channel tiles (rows y-1..y+1) into LDS with the
// Tensor Data Mover while all waves run V_WMMA_F32_16X16X4_F32 chains; the
// 16x16 product diagonals are the needed center-neighbor dots.
__global__ __launch_bounds__(256) void sim_wmma_kernel(const float* __restrict__ x_ema,
                                                       const float* __restrict__ x_src,
                                                       const float* __restrict__ norms,
                                                       float* __restrict__ sims) {
  const int y   = blockIdx.x;
  const int b   = blockIdx.y;
  const int inp = blockIdx.z;
  const float* src = (inp == 0 ? x_ema : x_src) + (size_t)b * CF * (HF * HF);

  __shared__ float lds[2 * CHUNK_WORDS];   // double buffer, 2 * 26112 B

  const int lane  = threadIdx.x & 31;
  const int wid   = threadIdx.x >> 5;      // 0..7
  const int x0    = wid * 16;
  const int mpx   = lane & 15;             // pixel within 16-block (A: M, B: N)
  const int chSel = (lane >> 4) * 2;       // K-pair select per WMMA layout

  int yClamp = y - 1;
  if (yClamp < 0) yClamp = 0;
  if (yClamp > HF - 3) yClamp = HF - 3;
  const int rC = y - yClamp;               // center row index inside staged tile

  unsigned lds_base =
      (unsigned)(uintptr_t)(__attribute__((address_space(3))) const float*)lds;

  v8f zero = {0.f, 0.f, 0.f, 0.f, 0.f, 0.f, 0.f, 0.f};
  v8f acc[9];
#pragma unroll
  for (int d = 0; d < 9; ++d) acc[d] = zero;

  if (wid == 0)
    tdm_load_chunk(src + (size_t)yClamp * HF, lds_base);

  for (int chunk = 0; chunk < CF / TDM_CH; ++chunk) {
    if (wid == 0) {
      if (chunk + 1 < CF / TDM_CH) {
        tdm_load_chunk(src + (size_t)(chunk + 1) * TDM_CH * (HF * HF) + (size_t)yClamp * HF,
                       lds_base + (unsigned)(((chunk + 1) & 1) * CHUNK_WORDS * 4));
        __builtin_amdgcn_s_wait_tensorcnt(1);   // current chunk landed
      } else {
        __builtin_amdgcn_s_wait_tensorcnt(0);
      }
    }
    __syncthreads();
    const float* buf = &lds[(chunk & 1) * CHUNK_WORDS];
#pragma unroll
    for (int ks = 0; ks < TDM_CH / 4; ++ks) {
      int cc = ks * 4 + chSel;               // channel within chunk for this lane half
      v2f a;
      a.x = buf[(cc * 3 + rC) * ROWSTR + x0 + mpx];
      a.y = buf[((cc + 1) * 3 + rC) * ROWSTR + x0 + mpx];
#pragma unroll
      for (int dh = 0; dh < 3; ++dh) {
        int py = y + dh - 1;
        int rB = py - yClamp;
#pragma unroll
        for (int dw = 0; dw < 3; ++dw) {
          int px = x0 + mpx + dw - 1;
          v2f bf = {0.0f, 0.0f};             // zero-pad OOB row/column
          if (py >= 0 && py < HF && px >= 0 && px < HF) {
            bf.x = buf[(cc * 3 + rB) * ROWSTR + px];
            bf.y = buf[((cc + 1) * 3 + rB) * ROWSTR + px];
          }
          acc[dh * 3 + dw] = __builtin_amdgcn_wmma_f32_16x16x4_f32(
              false, a, false, bf, (short)0, acc[dh * 3 + dw], false, false);
        }
      }
    }
    __syncthreads();
  }

  const float* nb = norms + ((size_t)(inp * HB + b)) * (HF * HF);
  float n_a = (lane < 16) ? nb[y * HF + x0 + mpx] : 0.0f;
  float* so = sims + ((size_t)(inp * HB + b) * 9) * (HF * HF);
#pragma unroll
  for (int dir = 0; dir < 9; ++dir) {
    const int dh = dir / 3 - 1;
    const int dw = dir % 3 - 1;
    // extract diag(m,m): m<8 -> lane m comp m ; m>=8 -> lane m+16 comp m-8
    float dval = 0.0f;
#pragma unroll
    for (int m = 0; m < 16; ++m) {
      float v = __shfl(acc[dir][m & 7], (m < 8) ? m : (m + 16), 32);
      if (lane == m) dval = v;
    }
    if (lane < 16) {
      int nx = x0 + mpx + dw;
      int ny = y + dh;
      float n_b = 0.0f;
      if (nx >= 0 && nx < HF && ny >= 0 && ny < HF) n_b = nb[ny * HF + nx];
      float dis = n_a + n_b - 2.0f * dval;   // OOB: dot=0, n_b=0 -> dis=||a||^2
      so[(size_t)dir * (HF * HF) + y * HF + x0 + mpx] = __expf(-dis * INV_SIG2);
    }
  }
}

// ---------------- fusion: masks + 4 masked sums per block ----------------
__global__ __launch_bounds__(256) void fusion_kernel(const int* __restrict__ gt_src,
                                                     const int* __restrict__ mix,
                                                     const float* __restrict__ prob,
                                                     const int* __restrict__ pred,
                                                     const float* __restrict__ sims,
                                                     float* __restrict__ partials) {
  int idx = blockIdx.x * 256 + threadIdx.x;        // exactly HB*HT*HT threads
  int b = idx >> 16;
  int pix = idx & 65535;
  int h = pix >> 8, w = pix & 255;

  float p[NC];
  const float* pc = prob + ((size_t)b * NC) * (HT * HT) + pix;
#pragma unroll
  for (int c = 0; c < NC; ++c) p[c] = pc[(size_t)c * (HT * HT)];
  int predc = pred[idx];
  int gtc = gt_src[(size_t)b * (1024 * 1024) + (h * 4) * 1024 + (w * 4)];
  bool ign_src_c = (gtc != 255);

  int y128 = h >> 1, x128 = w >> 1;
  float cross[9], emas[9];
  bool pospred[9];
  bool ign_full = true;
  float s_sp = 0, c_sp = 0, s_sn = 0, c_sn = 0;

#pragma unroll
  for (int k = 0; k < 9; ++k) {
    int dh = (k / 3 - 1) * 2, dw = (k % 3 - 1) * 2;
    int nh = h + dh, nw = w + dw;
    bool inb = (nh >= 0) && (nh < HT) && (nw >= 0) && (nw < HT);
    float cp = 0.0f;
    int gtn = 0, predn = 0;     // unfold zero-pad -> 0 when OOB
    bool ign_k = false;
    if (inb) {
      const float* pn = prob + ((size_t)b * NC) * (HT * HT) + nh * HT + nw;
      float d = 0.0f;
#pragma unroll
      for (int c = 0; c < NC; ++c) d += p[c] * pn[(size_t)c * (HT * HT)];
      cp = d;
      gtn = gt_src[(size_t)b * (1024 * 1024) + (nh * 4) * 1024 + (nw * 4)];
      predn = pred[(b << 16) + nh * HT + nw];
      int mm = mix[(size_t)b * (HT * HT) + nh * HT + nw];
      ign_k = (gtn != 255) && ((1.0f - (float)mm) > 0.5f);
    }
    ign_full = ign_full && ign_k;
    cross[k] = cp;
    pospred[k] = (predn == predc);
    float ss = sims[(((size_t)(1 * HB + b) * 9 + k)) * (HF * HF) + y128 * HF + x128];
    float es = sims[(((size_t)(0 * HB + b) * 9 + k)) * (HF * HF) + y128 * HF + x128];
    emas[k] = es;
    bool pg = (gtn == gtc) && ign_src_c;
    bool ng = (gtn != gtc) && ign_src_c;
    s_sp += pg ? ss : 0.0f;  c_sp += pg ? 1.0f : 0.0f;
    s_sn += ng ? ss : 0.0f;  c_sn += ng ? 1.0f : 0.0f;
  }

  float s_mp = 0, c_mp = 0, s_mn = 0, c_mn = 0;
#pragma unroll
  for (int k = 0; k < 9; ++k) {
    bool pm = (emas[k] < 0.25f) && pospred[k] && ign_full;
    bool nm = (emas[k] > 0.75f) && (!pospred[k]) && ign_full;
    s_mp += pm ? cross[k] : 0.0f;  c_mp += pm ? 1.0f : 0.0f;
    s_mn += nm ? cross[k] : 0.0f;  c_mn += nm ? 1.0f : 0.0f;
  }

  // deterministic block reduction
  __shared__ float red[8][8];
  float vals[8] = {s_sp, c_sp, s_sn, c_sn, s_mp, c_mp, s_mn, c_mn};
#pragma unroll
  for (int i = 0; i < 8; ++i) {
    float v = vals[i];
#pragma unroll
    for (int off = 16; off > 0; off >>= 1) v += __shfl_down(v, off, 32);
    if ((threadIdx.x & 31) == 0) red[threadIdx.x >> 5][i] = v;
  }
  __syncthreads();
  if (threadIdx.x == 0) {
#pragma unroll
    for (int i = 0; i < 8; ++i) {
      float s = 0.0f;
      for (int wv = 0; wv < 8; ++wv) s += red[wv][i];
      partials[blockIdx.x * 8 + i] = s;
    }
  }
}

// ---------------- finalize: ordered reduce + losses ----------------
__global__ void finalize_kernel(const float* __restrict__ partials, int nblk,
                                float* __restrict__ out) {
  if (threadIdx.x != 0 || blockIdx.x != 0) return;
  float a[8] = {0, 0, 0, 0, 0, 0, 0, 0};
  for (int bk = 0; bk < nblk; ++bk)
    for (int i = 0; i < 8; ++i) a[i] += partials[bk * 8 + i];
  float m_sp = a[0] / fmaxf(a[1], 1.0f);
  float m_sn = a[2] / fmaxf(a[3], 1.0f);
  float m_mp = a[4] / fmaxf(a[5], 1.0f);
  float m_mn = a[6] / fmaxf(a[7], 1.0f);
  out[0] = -m_sp;                               // * W_SRC_POS (=1)
  out[1] = m_sn;                                // * W_SRC_NEG (=1)
  out[2] = (a[5] > 0.0f) ? m_mp : 0.0f;         // * W_SIM_POS (=1)
  out[3] = (a[7] > 0.0f) ? -m_mn : 0.0f;        // * W_SIM_NEG (=1)
}

extern "C" void kernel_launch(void* const* d_in, const int* in_sizes, int n_in,
                              void* d_out, int out_size, void* d_ws, size_t ws_size,
                              hipStream_t stream) {
  (void)in_sizes; (void)n_in; (void)out_size; (void)ws_size;
  const float* logits_trg = (const float*)d_in[0];
  const int*   gt_src     = (const int*)d_in[2];
  const float* x_ema      = (const float*)d_in[3];
  const float* x_src      = (const float*)d_in[4];
  const int*   mix_masks  = (const int*)d_in[6];
  float* out = (float*)d_out;

  char* ws = (char*)d_ws;
  float* prob  = (float*)(ws);              // 2*19*65536 f32 = 9,961,472 B
  int*   pred  = (int*)(ws + 9961472);      // 2*65536 i32    =   524,288 B
  float* norms = (float*)(ws + 10485760);   // 2*2*16384 f32  =   262,144 B
  float* sims  = (float*)(ws + 10747904);   // 2*2*9*16384 f32= 2,359,296 B
  float* part  = (float*)(ws + 13107200);   // 512*8 f32      =    16,384 B

  prep_kernel<<<512, 256, 0, stream>>>(logits_trg, prob, pred);
  norm_kernel<<<dim3(HF, HB, 2), 128, 0, stream>>>(x_ema, x_src, norms);
  sim_wmma_kernel<<<dim3(HF, HB, 2), 256, 0, stream>>>(x_ema, x_src, norms, sims);
  fusion_kernel<<<512, 256, 0, stream>>>(gt_src, mix_masks, prob, pred, sims, part);
  finalize_kernel<<<1, 32, 0, stream>>>(part, 512, out);
}